// Model_79130477461984
// MI455X (gfx1250) — compile-verified
//
#include <hip/hip_runtime.h>
#include <math.h>

// Problem dims (fixed by the reference)
#define T_STEPS 8192
#define D_IN    128
#define HID     2048
#define ALPHA   0.1f

// Scan kernel config: 128 persistent WGs x 16 rows = 2048 rows of W_rec,
// each WG holds its 16x2048 fp32 slice (128KB) + full h (8KB) in LDS.
#define NBLK    128
#define ROWS    16

typedef __attribute__((ext_vector_type(2))) float v2f;
typedef __attribute__((ext_vector_type(8))) float v8f;

// ---------------------------------------------------------------------------
// Kernel 0: zero barrier counter, copy initial h into d_out row 0.
// ---------------------------------------------------------------------------
__global__ __launch_bounds__(256) void init_kernel(const float* __restrict__ h,
                                                   float* __restrict__ out,
                                                   unsigned* __restrict__ counter) {
    const int tid = threadIdx.x;
    if (tid == 0) *counter = 0u;
    const float4* src = (const float4*)h;
    float4* dst = (float4*)out;
    for (int i = tid; i < HID / 4; i += 256) dst[i] = src[i];
}

// ---------------------------------------------------------------------------
// Kernel 1: inp = u @ W_in^T via V_WMMA_F32_16X16X4_F32 (exact fp32).
// Writes inp[t, :] into d_out row (t+1); the scan consumes & overwrites it.
// M = T (8192), N = HID (2048), K = D_IN (128).
// Block = 256 threads = 8 waves; wave w owns one 16x16 tile along N.
// Grid  = (2048/(8*16), 8192/16) = (16, 512).
//
// fp32 WMMA VGPR layout (ISA 7.12.2):
//   A 16x4 : lane&15 = M row; VGPR0 = K{0|2}, VGPR1 = K{1|3} (half-wave split)
//   B 4x16 : lane&15 = N col; same K split
//   C/D    : lane&15 = N; VGPR r = M row r (+8 for lanes 16-31)
// ---------------------------------------------------------------------------
__global__ __launch_bounds__(256) void gemm_win_wmma(const float* __restrict__ u,
                                                     const float* __restrict__ Win,
                                                     float* __restrict__ out) {
    const int tid  = threadIdx.x;
    const int lane = tid & 31;
    const int w    = tid >> 5;
    const int n0   = (blockIdx.x * 8 + w) * 16;
    const int m0   = blockIdx.y * 16;
    const int mn   = lane & 15;          // M row for A, N col for B/C
    const int kb   = (lane >> 4) * 2;    // K sub-offset per half-wave

    const float* arow = u   + (size_t)(m0 + mn) * D_IN + kb;
    const float* brow = Win + (size_t)(n0 + mn) * D_IN + kb;  // B[k][n] = W_in[n][k]

    v8f c = {0.f, 0.f, 0.f, 0.f, 0.f, 0.f, 0.f, 0.f};
#pragma unroll
    for (int k0 = 0; k0 < D_IN; k0 += 4) {
        v2f a = *(const v2f*)(arow + k0);   // 8B-aligned: kb,k0 even
        v2f b = *(const v2f*)(brow + k0);
        c = __builtin_amdgcn_wmma_f32_16x16x4_f32(
                /*neg_a=*/false, a, /*neg_b=*/false, b,
                /*c_mod=*/(short)0, c, /*reuse_a=*/false, /*reuse_b=*/false);
    }

    const int mbase = m0 + (lane >> 4) * 8 + 1;  // +1: out row t+1 holds x_t
#pragma unroll
    for (int r = 0; r < 8; ++r)
        out[(size_t)(mbase + r) * HID + (n0 + mn)] = c[r];
}

// ---------------------------------------------------------------------------
// Kernel 2: persistent leaky-ESN scan. W_rec slice resident in LDS,
// device-wide atomic barrier per time step.
// Thread map: wave w, lane l -> local row i = 2w + (l>>4), chunk k = l&15.
// Each (i,k) accumulates 128 elements as 32 float4 MACs with stride-64
// chunking -> half-wave lanes hit disjoint 4-bank groups (conflict-free),
// then a 16-lane __shfl_xor tree reduces the row.
// ---------------------------------------------------------------------------
__global__ __launch_bounds__(256) void rnn_scan(const float* __restrict__ W_rec,
                                                float* __restrict__ out,
                                                unsigned* __restrict__ counter) {
    extern __shared__ float smem[];
    float* ws = smem;               // ROWS * HID  (128 KB)
    float* hb = smem + ROWS * HID;  // HID         (8 KB)

    const int tid  = threadIdx.x;
    const int b    = blockIdx.x;
    const int lane = tid & 31;
    const int wv   = tid >> 5;
    const int i    = wv * 2 + (lane >> 4);  // local row 0..15
    const int k    = lane & 15;             // chunk id 0..15
    const int gr   = b * ROWS + i;          // global row

    // Preload this WG's 16 rows of W_rec into LDS.
    {
        const float4* src = (const float4*)(W_rec + (size_t)b * ROWS * HID);
        float4* dst = (float4*)ws;
        for (int idx = tid; idx < ROWS * HID / 4; idx += 256) dst[idx] = src[idx];
    }
    __syncthreads();

    const float4* ws4 = (const float4*)(ws + i * HID);
    const float4* hb4 = (const float4*)hb;

    for (int t = 0; t < T_STEPS; ++t) {
        // Stage h_t (d_out row t) into LDS: 512 float4 over 256 threads.
        {
            const float4* src = (const float4*)(out + (size_t)t * HID);
            float4* dst = (float4*)hb;
            dst[tid]       = src[tid];
            dst[tid + 256] = src[tid + 256];
        }
        __syncthreads();

        // Row-i partial dot: elements c = cc*64 + 4k .. +3, cc = 0..31.
        float4 acc = {0.f, 0.f, 0.f, 0.f};
#pragma unroll
        for (int cc = 0; cc < 32; ++cc) {
            float4 wvv = ws4[cc * 16 + k];
            float4 hvv = hb4[cc * 16 + k];
            acc.x = fmaf(wvv.x, hvv.x, acc.x);
            acc.y = fmaf(wvv.y, hvv.y, acc.y);
            acc.z = fmaf(wvv.z, hvv.z, acc.z);
            acc.w = fmaf(wvv.w, hvv.w, acc.w);
        }
        float p = (acc.x + acc.y) + (acc.z + acc.w);
        p += __shfl_xor(p, 8, 16);
        p += __shfl_xor(p, 4, 16);
        p += __shfl_xor(p, 2, 16);
        p += __shfl_xor(p, 1, 16);

        if (k == 0) {
            const size_t oidx = (size_t)(t + 1) * HID + gr;
            float x    = out[oidx];        // precomputed input drive x_t[gr]
            float hold = hb[gr];
            float hn   = (1.0f - ALPHA) * hold + ALPHA * tanhf(p + x);
            out[oidx] = hn;
            __threadfence();               // make row visible device-wide
        }
        __syncthreads();

        // Device-wide step barrier: monotonic arrive counter.
        if (tid == 0) {
            __hip_atomic_fetch_add(counter, 1u, __ATOMIC_RELEASE,
                                   __HIP_MEMORY_SCOPE_AGENT);
            const unsigned tgt = (unsigned)(t + 1) * NBLK;
            while (__hip_atomic_load(counter, __ATOMIC_ACQUIRE,
                                     __HIP_MEMORY_SCOPE_AGENT) < tgt)
                __builtin_amdgcn_s_sleep(2);
        }
        __syncthreads();

        // Drop any stale h lines from WGP$ before the next row-t+1 read.
        asm volatile("global_inv scope:SCOPE_DEV\n\t"
                     "s_wait_loadcnt 0x0" ::: "memory");
    }
}

// ---------------------------------------------------------------------------
// inputs (setup_inputs order): u (T,128), h (2048), h_all (unused),
// W_in (2048,128), W_rec (2048,2048). d_out: (T+1, 2048) fp32.
// ---------------------------------------------------------------------------
extern "C" void kernel_launch(void* const* d_in, const int* in_sizes, int n_in,
                              void* d_out, int out_size, void* d_ws, size_t ws_size,
                              hipStream_t stream) {
    const float* u     = (const float*)d_in[0];
    const float* h     = (const float*)d_in[1];
    const float* W_in  = (const float*)d_in[3];
    const float* W_rec = (const float*)d_in[4];
    float* out = (float*)d_out;
    unsigned* counter = (unsigned*)d_ws;

    init_kernel<<<1, 256, 0, stream>>>(h, out, counter);

    dim3 g1(HID / (8 * 16), T_STEPS / 16);  // (16, 512)
    gemm_win_wmma<<<g1, 256, 0, stream>>>(u, W_in, out);

    const size_t lds_bytes = (size_t)(ROWS * HID + HID) * sizeof(float);  // 136 KB
    rnn_scan<<<NBLK, 256, lds_bytes, stream>>>(W_rec, out, counter);
}